// GraphEmbedding_68453188763767
// MI455X (gfx1250) — compile-verified
//
#include <hip/hip_runtime.h>
#include <math.h>

// ---------------------------------------------------------------------------
// TGN graph embedding, fused for MI455X (gfx1250, wave32, WMMA).
//
// Algebraic restructure: mask-weighted mean of affine messages
//   == (mask-weighted SUM of raw features) @ W + [cnt>0]*bias terms,
// so we aggregate raw 172/172/100-dim features per node first, then run ONE
// fused GEMM per node row against stacked weights Wcat[448x128]
// (rows 0..171 = W_node, 172..343 = W_edge, 344..443 = W_time, 444..447 = 0).
// This cuts GEMM FLOPs 20x; remaining cost is ~2.3GB of 688B feature gathers
// (~100us @ 23.3 TB/s HBM). GEMM kept in full f32 via V_WMMA_F32_16X16X4_F32
// (exact w.r.t. the f32 reference; matrix work is far from the bottleneck).
// ---------------------------------------------------------------------------

typedef float v2f __attribute__((ext_vector_type(2)));
typedef float v8f __attribute__((ext_vector_type(8)));

#define KNBR   20
#define DF     172     // node/edge feature dim
#define DTF    100     // time feature dim
#define DEMB   128     // embedding dim
#define KDIM   448     // 172 + 172 + 100 padded to multiple of 16
#define KD_E0  172
#define KD_T0  344
#define LDSTR  452     // LDS row stride (floats): 452 mod 64 == 4 -> the 4
                       // consecutive banks per row (4M+kb..+3) tile all 64
                       // banks across M=0..15 -> conflict-free A-frag loads.

// ---------------------------------------------------------------------------
// Build stacked weight matrix Wcat[448][128].  <<<448, 128>>>
// ---------------------------------------------------------------------------
__global__ void build_wcat(const float* __restrict__ Wn,
                           const float* __restrict__ We,
                           const float* __restrict__ Wt,
                           float* __restrict__ wcat) {
  const int r = blockIdx.x;
  const int n = threadIdx.x;
  float v = 0.0f;
  if (r < DF)            v = Wn[r * DEMB + n];
  else if (r < 2 * DF)   v = We[(r - DF) * DEMB + n];
  else if (r < KDIM - 4) v = Wt[(r - 2 * DF) * DEMB + n];
  wcat[r * DEMB + n] = v;
}

// ---------------------------------------------------------------------------
// tz = time_tr(0) = cos(phase) @ W_time + b_time  (constant 128-vector).
// <<<1, 128>>>
// ---------------------------------------------------------------------------
__global__ void build_tz(const float* __restrict__ Wt,
                         const float* __restrict__ bt,
                         const float* __restrict__ phase,
                         float* __restrict__ tz) {
  const int n = threadIdx.x;
  float acc = bt[n];
  for (int i = 0; i < DTF; ++i)
    acc += cosf(phase[i]) * Wt[i * DEMB + n];
  tz[n] = acc;
}

// ---------------------------------------------------------------------------
// Shared WMMA phase: 16 aggregated rows live in LDS (Xt), each of the 8 waves
// owns one 16-wide column tile of the 128 output columns and accumulates
// 112 k-steps of V_WMMA_F32_16X16X4_F32.
// A frag (16x4 f32): lanes 0-15 -> M=lane, k={kb,kb+1}; lanes 16-31 -> k+2,k+3.
// B frag (4x16 f32): VGPR0 = rows kb / kb+2, VGPR1 = rows kb+1 / kb+3.
// ---------------------------------------------------------------------------
__device__ __forceinline__ v8f wmma_tile(const float* __restrict__ Xt,
                                         const float* __restrict__ wcat,
                                         int lane, int n0) {
  const int  Mrow  = lane & 15;
  const int  khalf = (lane >> 4) * 2;
  const int  ncol  = n0 + Mrow;   // B/D column owned by this lane
  v8f acc = {};
#pragma unroll 4
  for (int kb = 0; kb < KDIM; kb += 4) {
    const float* ap = &Xt[Mrow * LDSTR + kb + khalf];
    v2f a; a.x = ap[0]; a.y = ap[1];
    v2f b;
    b.x = wcat[(kb + khalf)     * DEMB + ncol];
    b.y = wcat[(kb + khalf + 1) * DEMB + ncol];
    acc = __builtin_amdgcn_wmma_f32_16x16x4_f32(false, a, false, b,
                                                (short)0, acc, false, false);
  }
  return acc;
}

// ---------------------------------------------------------------------------
// Stage A: emb1[m] for m in [0, B*K).  Block = 256 thr (8 waves) = 16 rows.
// Phase 1: each wave aggregates 2 rows of raw features into LDS X tile.
// Phase 2: WMMA 16x128 tile, epilogue adds b_node + tz + flag*(bn+be+bt).
// <<<81920/16, 256>>>
// ---------------------------------------------------------------------------
__global__ __launch_bounds__(256)
void stageA(const float* __restrict__ nodeF, const float* __restrict__ edgeF,
            const float* __restrict__ wcat,  const float* __restrict__ tz,
            const float* __restrict__ b_node, const float* __restrict__ b_edge,
            const float* __restrict__ b_time,
            const float* __restrict__ time_w, const float* __restrict__ time_ph,
            const int*   __restrict__ nbrs_l2_flat,
            const float* __restrict__ timestamps,
            const int*   __restrict__ nbrs_l1, const int* __restrict__ eidx_l1,
            const float* __restrict__ etime_l1,
            float* __restrict__ emb1) {
  __shared__ float Xt[16 * LDSTR];
  __shared__ float flags[16];

  const int tid  = threadIdx.x;
  const int wave = tid >> 5;
  const int lane = tid & 31;
  const int m0   = blockIdx.x * 16;

  // ---- phase 1: masked raw-feature aggregation, 2 rows per wave ----
  for (int rr = 0; rr < 2; ++rr) {
    const int   ml = wave * 2 + rr;
    const int   m  = m0 + ml;
    const float ts = timestamps[m / KNBR];           // repeat(timestamps, K)
    float accN[6] = {0,0,0,0,0,0};
    float accE[6] = {0,0,0,0,0,0};
    float accC[4] = {0,0,0,0};
    int cnt = 0;
    for (int j = 0; j < KNBR; ++j) {
      const int nbr = nbrs_l1[m * KNBR + j];
      if (nbr != 0) {
        ++cnt;
        const long nb = (long)nbr * DF;
        const long eb = (long)eidx_l1[m * KNBR + j] * DF;
#pragma unroll
        for (int c = 0; c < 6; ++c) {
          const int f = lane + 32 * c;
          if (f < DF) { accN[c] += nodeF[nb + f]; accE[c] += edgeF[eb + f]; }
        }
        const float dt = ts - etime_l1[m * KNBR + j];
#pragma unroll
        for (int c = 0; c < 4; ++c) {
          const int f = lane + 32 * c;
          if (f < DTF) accC[c] += cosf(dt * time_w[f] + time_ph[f]);
        }
      }
    }
    const float inv = 1.0f / (float)(cnt > 0 ? cnt : 1);
    const long  sb  = (long)nbrs_l2_flat[m] * DF;    // layer-1 "src" node
    float* X = &Xt[ml * LDSTR];
#pragma unroll
    for (int c = 0; c < 6; ++c) {
      const int f = lane + 32 * c;
      if (f < DF) {
        X[f]         = nodeF[sb + f] + inv * accN[c];  // src + agg (same W)
        X[KD_E0 + f] = inv * accE[c];
      }
    }
#pragma unroll
    for (int c = 0; c < 4; ++c) {
      const int f = lane + 32 * c;
      if (f < DTF) X[KD_T0 + f] = inv * accC[c];
    }
    if (lane < 4)  X[KDIM - 4 + lane] = 0.0f;        // zero pad rows 444..447
    if (lane == 0) flags[ml] = (cnt > 0) ? 1.0f : 0.0f;
  }
  __syncthreads();

  // ---- phase 2: WMMA + epilogue ----
  const int n0   = wave * 16;
  const int ncol = n0 + (lane & 15);
  v8f acc = wmma_tile(Xt, wcat, lane, n0);
  const float bias = b_node[ncol] + tz[ncol];
  const float bsum = b_node[ncol] + b_edge[ncol] + b_time[ncol];
  const int   mhi  = (lane >> 4) * 8;
#pragma unroll
  for (int r = 0; r < 8; ++r) {
    const int M = r + mhi;
    emb1[(long)(m0 + M) * DEMB + ncol] = acc[r] + bias + flags[M] * bsum;
  }
}

// ---------------------------------------------------------------------------
// Stage B: emb2[b] for b in [0, 4096).  Same structure; neighbor embeddings
// (emb1) are already in D-space so their masked mean is added in the epilogue.
// <<<4096/16, 256>>>
// ---------------------------------------------------------------------------
__global__ __launch_bounds__(256)
void stageB(const float* __restrict__ nodeF, const float* __restrict__ edgeF,
            const float* __restrict__ wcat,  const float* __restrict__ tz,
            const float* __restrict__ b_node, const float* __restrict__ b_edge,
            const float* __restrict__ b_time,
            const float* __restrict__ time_w, const float* __restrict__ time_ph,
            const int*   __restrict__ source_nodes,
            const float* __restrict__ timestamps,
            const int*   __restrict__ nbrs_l2, const int* __restrict__ eidx_l2,
            const float* __restrict__ etime_l2,
            const float* __restrict__ emb1,
            float* __restrict__ out) {
  __shared__ float Xt[16 * LDSTR];
  __shared__ float flags[16];
  __shared__ float Dagg[16][DEMB];   // inv * sum_k valid * emb1[b,k,:]

  const int tid  = threadIdx.x;
  const int wave = tid >> 5;
  const int lane = tid & 31;
  const int b0   = blockIdx.x * 16;

  for (int rr = 0; rr < 2; ++rr) {
    const int   ml = wave * 2 + rr;
    const int   b  = b0 + ml;
    const float ts = timestamps[b];
    float accE[6] = {0,0,0,0,0,0};
    float accC[4] = {0,0,0,0};
    float accD[4] = {0,0,0,0};
    int cnt = 0;
    for (int j = 0; j < KNBR; ++j) {
      const int nbr = nbrs_l2[b * KNBR + j];
      if (nbr != 0) {
        ++cnt;
        const long eb = (long)eidx_l2[b * KNBR + j] * DF;
#pragma unroll
        for (int c = 0; c < 6; ++c) {
          const int f = lane + 32 * c;
          if (f < DF) accE[c] += edgeF[eb + f];
        }
        const float dt = ts - etime_l2[b * KNBR + j];
#pragma unroll
        for (int c = 0; c < 4; ++c) {
          const int f = lane + 32 * c;
          if (f < DTF) accC[c] += cosf(dt * time_w[f] + time_ph[f]);
        }
        const float* ep = &emb1[(long)(b * KNBR + j) * DEMB];
#pragma unroll
        for (int c = 0; c < 4; ++c) accD[c] += ep[lane + 32 * c];
      }
    }
    const float inv = 1.0f / (float)(cnt > 0 ? cnt : 1);
    const long  sb  = (long)source_nodes[b] * DF;
    float* X = &Xt[ml * LDSTR];
#pragma unroll
    for (int c = 0; c < 6; ++c) {
      const int f = lane + 32 * c;
      if (f < DF) {
        X[f]         = nodeF[sb + f];
        X[KD_E0 + f] = inv * accE[c];
      }
    }
#pragma unroll
    for (int c = 0; c < 4; ++c) {
      const int f = lane + 32 * c;
      if (f < DTF) X[KD_T0 + f] = inv * accC[c];
    }
#pragma unroll
    for (int c = 0; c < 4; ++c) Dagg[ml][lane + 32 * c] = inv * accD[c];
    if (lane < 4)  X[KDIM - 4 + lane] = 0.0f;
    if (lane == 0) flags[ml] = (cnt > 0) ? 1.0f : 0.0f;
  }
  __syncthreads();

  const int n0   = wave * 16;
  const int ncol = n0 + (lane & 15);
  v8f acc = wmma_tile(Xt, wcat, lane, n0);
  const float bias = b_node[ncol] + tz[ncol];
  const float bsum = b_edge[ncol] + b_time[ncol];   // nbr_emb carries no bias
  const int   mhi  = (lane >> 4) * 8;
#pragma unroll
  for (int r = 0; r < 8; ++r) {
    const int M = r + mhi;
    out[(long)(b0 + M) * DEMB + ncol] =
        acc[r] + bias + flags[M] * bsum + Dagg[M][ncol];
  }
}

// ---------------------------------------------------------------------------
extern "C" void kernel_launch(void* const* d_in, const int* in_sizes, int n_in,
                              void* d_out, int out_size, void* d_ws,
                              size_t ws_size, hipStream_t stream) {
  const float* node_features = (const float*)d_in[0];
  const float* edge_features = (const float*)d_in[1];
  const float* W_node        = (const float*)d_in[2];
  const float* b_node        = (const float*)d_in[3];
  const float* W_time        = (const float*)d_in[4];
  const float* b_time        = (const float*)d_in[5];
  const float* W_edge        = (const float*)d_in[6];
  const float* b_edge        = (const float*)d_in[7];
  const float* time_w        = (const float*)d_in[8];
  const float* time_phase    = (const float*)d_in[9];
  const int*   source_nodes  = (const int*)  d_in[10];
  const float* timestamps    = (const float*)d_in[11];
  const int*   nbrs_l2       = (const int*)  d_in[12];
  const int*   eidx_l2       = (const int*)  d_in[13];
  const float* etime_l2      = (const float*)d_in[14];
  const int*   nbrs_l1       = (const int*)  d_in[15];
  const int*   eidx_l1       = (const int*)  d_in[16];
  const float* etime_l1      = (const float*)d_in[17];

  const int Bn = in_sizes[10];          // 4096
  const int M1 = Bn * KNBR;             // 81920 layer-1 rows

  // workspace layout (floats): Wcat[448*128] | tz[128] | emb1[M1*128]
  float* ws   = (float*)d_ws;
  float* wcat = ws;
  float* tzv  = ws + KDIM * DEMB;
  float* emb1 = ws + KDIM * DEMB + DEMB;

  build_wcat<<<KDIM, DEMB, 0, stream>>>(W_node, W_edge, W_time, wcat);
  build_tz<<<1, DEMB, 0, stream>>>(W_time, b_time, time_phase, tzv);

  stageA<<<M1 / 16, 256, 0, stream>>>(
      node_features, edge_features, wcat, tzv, b_node, b_edge, b_time,
      time_w, time_phase, nbrs_l2 /* flattened = layer-1 src ids */,
      timestamps, nbrs_l1, eidx_l1, etime_l1, emb1);

  stageB<<<Bn / 16, 256, 0, stream>>>(
      node_features, edge_features, wcat, tzv, b_node, b_edge, b_time,
      time_w, time_phase, source_nodes, timestamps,
      nbrs_l2, eidx_l2, etime_l2, emb1, (float*)d_out);
}